// AnchorEncoder_40604620816501
// MI455X (gfx1250) — compile-verified
//
#include <hip/hip_runtime.h>

// ---- problem constants (match reference) ----
constexpr int Bc   = 2;
constexpr int Lc   = 512;
constexpr int Hc   = 64;     // H == C == 64
constexpr int Kk   = 16;
constexpr int Nn   = 300;
constexpr int N1c  = 301;
constexpr int N1Pc = 304;    // N1 padded to multiple of 16 (K-tiling, zero pad)

typedef __attribute__((ext_vector_type(2))) float v2f;
typedef __attribute__((ext_vector_type(8))) float v8f;

__device__ inline v8f wmma4(v2f a, v2f b, v8f c) {
  // V_WMMA_F32_16X16X4_F32 : D = A(16x4) * B(4x16) + C(16x16)
  return __builtin_amdgcn_wmma_f32_16x16x4_f32(
      /*neg_a=*/false, a, /*neg_b=*/false, b,
      /*c_mod=*/(short)0, c, /*reuse_a=*/false, /*reuse_b=*/false);
}

__device__ inline v8f splat8(float x) {
  v8f v = {x, x, x, x, x, x, x, x};
  return v;
}

// ---------------------------------------------------------------------------
// sim[b,i,j] = dot(nf[b,i,:], nf[b,j,:]); zero (tgt,-1) and (-1,tgt)
// grid: B*N1 blocks, 512 threads
// ---------------------------------------------------------------------------
__global__ void sim_kernel(const float* __restrict__ nf,
                           const int* __restrict__ tgtp,
                           float* __restrict__ sim) {
  __shared__ float nfi[Hc];
  const int bi = blockIdx.x;
  const int b = bi / N1c, i = bi % N1c;
  const int tid = threadIdx.x;
  const float* base = nf + (size_t)b * N1c * Hc;
  if (tid < Hc) nfi[tid] = base[i * Hc + tid];
  __syncthreads();
  const int j = tid;
  if (j < N1c) {
    const float* nfj = base + j * Hc;
    float s = 0.f;
#pragma unroll 8
    for (int c = 0; c < Hc; ++c) s += nfi[c] * nfj[c];
    const int tg = *tgtp;
    if ((i == tg && j == N1c - 1) || (i == N1c - 1 && j == tg)) s = 0.f;
    sim[((size_t)b * N1c + i) * N1c + j] = s;
  }
}

// ---------------------------------------------------------------------------
// top-K mask over sim[b, N1-1, 0:300] -> nm1[b, 0:301] (last entry forced 1)
// grid: B blocks; serial (tiny)
// ---------------------------------------------------------------------------
__global__ void topk_kernel(const float* __restrict__ sim,
                            float* __restrict__ mask) {
  const int b = blockIdx.x;
  if (threadIdx.x != 0) return;
  const float* row = sim + ((size_t)b * N1c + (N1c - 1)) * N1c;
  float vals[Nn];
  for (int j = 0; j < Nn; ++j) vals[j] = row[j];
  float* mk = mask + b * N1c;
  for (int j = 0; j < N1c; ++j) mk[j] = 0.f;
  mk[N1c - 1] = 1.f;
  for (int t = 0; t < Kk; ++t) {
    float best = -3.4e38f;
    int arg = 0;
    for (int j = 0; j < Nn; ++j)
      if (vals[j] > best) { best = vals[j]; arg = j; }
    mk[arg] = 1.f;
    vals[arg] = -3.4e38f;
  }
}

// ---------------------------------------------------------------------------
// degree -> d^{-1/2} for both global and local adjacency (with +I)
// grid: B*N1 blocks, 512 threads
// ---------------------------------------------------------------------------
__global__ void deg_kernel(const float* __restrict__ sim,
                           const float* __restrict__ mask,
                           float* __restrict__ dg, float* __restrict__ dl) {
  __shared__ float rg[512];
  __shared__ float rl[512];
  const int bi = blockIdx.x;
  const int b = bi / N1c, i = bi % N1c;
  const int tid = threadIdx.x;
  float sg = 0.f, sl = 0.f;
  if (tid < N1c) {
    const float s = sim[((size_t)b * N1c + i) * N1c + tid];
    const float diag = (tid == i) ? 1.f : 0.f;
    sg = s + diag;
    sl = s * mask[b * N1c + i] * mask[b * N1c + tid] + diag;
  }
  rg[tid] = sg;
  rl[tid] = sl;
  __syncthreads();
  for (int s = 256; s > 0; s >>= 1) {
    if (tid < s) { rg[tid] += rg[tid + s]; rl[tid] += rl[tid + s]; }
    __syncthreads();
  }
  if (tid == 0) {
    const float a = rg[0], c = rl[0];
    dg[b * N1c + i] = (a > 0.f) ? rsqrtf(fmaxf(a, 1e-12f)) : 0.f;
    dl[b * N1c + i] = (c > 0.f) ? rsqrtf(fmaxf(c, 1e-12f)) : 0.f;
  }
}

// ---------------------------------------------------------------------------
// write zero-padded normalized adjacencies: [B][304][304]
// ---------------------------------------------------------------------------
__global__ void gso_kernel(const float* __restrict__ sim,
                           const float* __restrict__ mask,
                           const float* __restrict__ dg,
                           const float* __restrict__ dl,
                           float* __restrict__ gsoG, float* __restrict__ gsoL) {
  const int idx = blockIdx.x * blockDim.x + threadIdx.x;
  const int total = Bc * N1Pc * N1Pc;
  if (idx >= total) return;
  const int b = idx / (N1Pc * N1Pc);
  const int rem = idx % (N1Pc * N1Pc);
  const int i = rem / N1Pc, j = rem % N1Pc;
  float g = 0.f, l = 0.f;
  if (i < N1c && j < N1c) {
    const float s = sim[((size_t)b * N1c + i) * N1c + j];
    const float diag = (i == j) ? 1.f : 0.f;
    g = dg[b * N1c + i] * (s + diag) * dg[b * N1c + j];
    l = dl[b * N1c + i] *
        (s * mask[b * N1c + i] * mask[b * N1c + j] + diag) * dl[b * N1c + j];
  }
  gsoG[idx] = g;
  gsoL[idx] = l;
}

// ---------------------------------------------------------------------------
// Fused GCN layer:  h_out = relu( (gso @ h_in) @ W + bias ) + h_in
// one block per (b,l); 8 waves; h_in[b,l] staged in LDS (zero-padded rows).
// WMMA f32 16x16x4 for both GEMMs.
// ---------------------------------------------------------------------------
__global__ __launch_bounds__(256)
void gcn_layer_kernel(const float* __restrict__ gso,   // [B][304][304]
                      const float* __restrict__ h_in,  // [B][L][301][64]
                      const float* __restrict__ W,     // [64][64]
                      const float* __restrict__ bias,  // [64]
                      float* __restrict__ h_out) {
  __shared__ float hS[N1Pc * Hc];     // 77824 B
  __shared__ float aggS[8 * 16 * Hc]; // 32768 B (wave-private tiles)
  const int tid = threadIdx.x;
  const int bl = blockIdx.x;
  const int b = bl >> 9;   // L == 512
  const int l = bl & 511;
  const float* hin = h_in + ((size_t)(b * Lc + l)) * N1c * Hc;

  for (int idx = tid; idx < (N1Pc * Hc) / 4; idx += 256) {
    const int e = idx * 4;
    const int row = e >> 6;
    const int col = e & 63;
    float4 v = make_float4(0.f, 0.f, 0.f, 0.f);
    if (row < N1c) v = *(const float4*)(hin + row * Hc + col);
    *(float4*)(hS + e) = v;
  }
  __syncthreads();

  const int wave = tid >> 5, lane = tid & 31;
  const int m = lane & 15, kh = lane >> 4;
  const float* gso_b = gso + (size_t)b * N1Pc * N1Pc;
  float* hout = h_out + ((size_t)(b * Lc + l)) * N1c * Hc;

  for (int it = wave; it < N1Pc / 16; it += 8) {
    const int i0 = it * 16;
    v8f acc[4] = {};
    const float* grow = gso_b + (size_t)(i0 + m) * N1Pc + 2 * kh;
    for (int kk = 0; kk < N1Pc; kk += 4) {
      const v2f a = *(const v2f*)(grow + kk);
      const float* hb = hS + (kk + 2 * kh) * Hc + m;
#pragma unroll
      for (int n = 0; n < 4; ++n) {
        v2f bb;
        bb.x = hb[n * 16];
        bb.y = hb[n * 16 + Hc];
        acc[n] = wmma4(a, bb, acc[n]);
      }
    }
    // spill agg tile to wave-private LDS to re-form A fragments
    float* aS = aggS + wave * 16 * Hc;
#pragma unroll
    for (int n = 0; n < 4; ++n)
#pragma unroll
      for (int r = 0; r < 8; ++r)
        aS[(r + 8 * kh) * Hc + n * 16 + m] = acc[n][r];

    v8f acc2[4];
#pragma unroll
    for (int n = 0; n < 4; ++n) acc2[n] = splat8(bias[n * 16 + m]);

    for (int kk = 0; kk < Hc; kk += 4) {
      const v2f a2 = *(const v2f*)(aS + m * Hc + kk + 2 * kh);
      const float* wb = W + (kk + 2 * kh) * Hc + m;
#pragma unroll
      for (int n = 0; n < 4; ++n) {
        v2f bb;
        bb.x = wb[n * 16];
        bb.y = wb[n * 16 + Hc];
        acc2[n] = wmma4(a2, bb, acc2[n]);
      }
    }
#pragma unroll
    for (int n = 0; n < 4; ++n)
#pragma unroll
      for (int r = 0; r < 8; ++r) {
        const int row = i0 + r + 8 * kh;
        if (row < N1c) {
          const int col = n * 16 + m;
          float v = fmaxf(acc2[n][r], 0.f);
          hout[row * Hc + col] = v + hS[row * Hc + col];  // residual
        }
      }
  }
}

// ---------------------------------------------------------------------------
// gMLP: gm = relu([ga|la] @ W1 + b1) @ W2 + b2 over R = B*L*N1 rows
// block = 128 rows (8 waves x 16); R divisible by 128.
// ---------------------------------------------------------------------------
__global__ __launch_bounds__(256)
void gmlp_kernel(const float* __restrict__ ga, const float* __restrict__ la,
                 const float* __restrict__ w1, const float* __restrict__ b1,
                 const float* __restrict__ w2, const float* __restrict__ b2,
                 float* __restrict__ gm) {
  __shared__ float zS[8 * 16 * Hc];
  const int tid = threadIdx.x;
  const int wave = tid >> 5, lane = tid & 31;
  const int m = lane & 15, kh = lane >> 4;
  const size_t r0 = (size_t)blockIdx.x * 128 + wave * 16;

  v8f acc[4];
#pragma unroll
  for (int n = 0; n < 4; ++n) acc[n] = splat8(b1[n * 16 + m]);

  // ga part (W1 rows 0..63)
  {
    const float* arow = ga + (r0 + m) * Hc + 2 * kh;
    for (int kk = 0; kk < Hc; kk += 4) {
      const v2f a = *(const v2f*)(arow + kk);
      const float* wb = w1 + (kk + 2 * kh) * Hc + m;
#pragma unroll
      for (int n = 0; n < 4; ++n) {
        v2f bb;
        bb.x = wb[n * 16];
        bb.y = wb[n * 16 + Hc];
        acc[n] = wmma4(a, bb, acc[n]);
      }
    }
  }
  // la part (W1 rows 64..127)
  {
    const float* arow = la + (r0 + m) * Hc + 2 * kh;
    for (int kk = 0; kk < Hc; kk += 4) {
      const v2f a = *(const v2f*)(arow + kk);
      const float* wb = w1 + (Hc + kk + 2 * kh) * Hc + m;
#pragma unroll
      for (int n = 0; n < 4; ++n) {
        v2f bb;
        bb.x = wb[n * 16];
        bb.y = wb[n * 16 + Hc];
        acc[n] = wmma4(a, bb, acc[n]);
      }
    }
  }
  float* zs = zS + wave * 16 * Hc;
#pragma unroll
  for (int n = 0; n < 4; ++n)
#pragma unroll
    for (int r = 0; r < 8; ++r)
      zs[(r + 8 * kh) * Hc + n * 16 + m] = fmaxf(acc[n][r], 0.f);

  v8f acc2[4];
#pragma unroll
  for (int n = 0; n < 4; ++n) acc2[n] = splat8(b2[n * 16 + m]);
  for (int kk = 0; kk < Hc; kk += 4) {
    const v2f a2 = *(const v2f*)(zs + m * Hc + kk + 2 * kh);
    const float* wb = w2 + (kk + 2 * kh) * Hc + m;
#pragma unroll
    for (int n = 0; n < 4; ++n) {
      v2f bb;
      bb.x = wb[n * 16];
      bb.y = wb[n * 16 + Hc];
      acc2[n] = wmma4(a2, bb, acc2[n]);
    }
  }
  float* out = gm + r0 * Hc;
#pragma unroll
  for (int n = 0; n < 4; ++n)
#pragma unroll
    for (int r = 0; r < 8; ++r)
      out[(r + 8 * kh) * Hc + n * 16 + m] = acc2[n][r];
}

// ---------------------------------------------------------------------------
// node MLP: y = relu(gm^T @ nmlp_w1 + b1); out[b,l,c] = y[c,:] . w2 + b2
// one block per (b,l); gm[b,l] staged in LDS (padded); deterministic reduce.
// ---------------------------------------------------------------------------
__global__ __launch_bounds__(256)
void nmlp_kernel(const float* __restrict__ gm,
                 const float* __restrict__ w1,   // [301][64]
                 const float* __restrict__ b1,   // [64]
                 const float* __restrict__ w2,   // [64]
                 const float* __restrict__ b2,   // [1]
                 float* __restrict__ out) {
  __shared__ float gS[N1Pc * Hc];  // 77824 B
  __shared__ float pS[16 * 16];    // per-(mt,nt)-tile row partials
  const int tid = threadIdx.x;
  const int bl = blockIdx.x;  // b*L + l
  const float* g = gm + (size_t)bl * N1c * Hc;

  for (int idx = tid; idx < (N1Pc * Hc) / 4; idx += 256) {
    const int e = idx * 4;
    const int row = e >> 6;
    const int col = e & 63;
    float4 v = make_float4(0.f, 0.f, 0.f, 0.f);
    if (row < N1c) v = *(const float4*)(g + row * Hc + col);
    *(float4*)(gS + e) = v;
  }
  __syncthreads();

  const int wave = tid >> 5, lane = tid & 31;
  const int m = lane & 15, kh = lane >> 4;

  for (int t = wave; t < 16; t += 8) {
    const int mt = t >> 2, nt = t & 3;
    const int c0 = mt * 16, h0 = nt * 16;
    v8f acc = splat8(b1[h0 + m]);
    for (int kk = 0; kk < N1Pc; kk += 4) {
      const int krow = kk + 2 * kh;
      v2f a;
      a.x = gS[krow * Hc + c0 + m];
      a.y = gS[(krow + 1) * Hc + c0 + m];
      v2f bb;
      bb.x = (krow < N1c) ? w1[krow * Hc + h0 + m] : 0.f;
      bb.y = (krow + 1 < N1c) ? w1[(krow + 1) * Hc + h0 + m] : 0.f;
      acc = wmma4(a, bb, acc);
    }
    const float wv = w2[h0 + m];
    float p[8];
#pragma unroll
    for (int r = 0; r < 8; ++r) p[r] = fmaxf(acc[r], 0.f) * wv;
    // reduce over the 16 lanes that share a row (each half independently)
#pragma unroll
    for (int off = 1; off < 16; off <<= 1)
#pragma unroll
      for (int r = 0; r < 8; ++r) p[r] += __shfl_xor(p[r], off, 16);
    if (m == 0) {
#pragma unroll
      for (int r = 0; r < 8; ++r) pS[t * 16 + (r + 8 * kh)] = p[r];
    }
  }
  __syncthreads();
  if (tid < Hc) {
    const int mt = tid >> 4;
    const int rl = tid & 15;
    float s = b2[0];
#pragma unroll
    for (int nt = 0; nt < 4; ++nt) s += pS[(mt * 4 + nt) * 16 + rl];
    out[(size_t)bl * Hc + tid] = s;
  }
}

// ---------------------------------------------------------------------------
extern "C" void kernel_launch(void* const* d_in, const int* in_sizes, int n_in,
                              void* d_out, int out_size, void* d_ws,
                              size_t ws_size, hipStream_t stream) {
  (void)in_sizes; (void)n_in; (void)out_size; (void)ws_size;
  const float* nf       = (const float*)d_in[0];
  const float* all_nodes= (const float*)d_in[1];
  const float* gcn_w0   = (const float*)d_in[2];
  const float* gcn_b0   = (const float*)d_in[3];
  const float* gcn_w1   = (const float*)d_in[4];
  const float* gcn_b1   = (const float*)d_in[5];
  const float* gmlp_w1  = (const float*)d_in[6];
  const float* gmlp_b1  = (const float*)d_in[7];
  const float* gmlp_w2  = (const float*)d_in[8];
  const float* gmlp_b2  = (const float*)d_in[9];
  const float* nmlp_w1  = (const float*)d_in[10];
  const float* nmlp_b1  = (const float*)d_in[11];
  const float* nmlp_w2  = (const float*)d_in[12];
  const float* nmlp_b2  = (const float*)d_in[13];
  const int*   tgt      = (const int*)d_in[14];

  float* ws = (float*)d_ws;
  size_t off = 0;
  float* sim  = ws + off; off += (size_t)Bc * N1c * N1c;
  float* mask = ws + off; off += (size_t)Bc * N1c;
  float* dg   = ws + off; off += (size_t)Bc * N1c;
  float* dl   = ws + off; off += (size_t)Bc * N1c;
  float* gsoG = ws + off; off += (size_t)Bc * N1Pc * N1Pc;
  float* gsoL = ws + off; off += (size_t)Bc * N1Pc * N1Pc;
  const size_t BIG = (size_t)Bc * Lc * N1c * Hc;
  float* bufT  = ws + off; off += BIG;   // h1 scratch, later gm
  float* bufGA = ws + off; off += BIG;   // ga
  float* bufLA = ws + off; off += BIG;   // la
  float* outp = (float*)d_out;

  sim_kernel<<<Bc * N1c, 512, 0, stream>>>(nf, tgt, sim);
  topk_kernel<<<Bc, 32, 0, stream>>>(sim, mask);
  deg_kernel<<<Bc * N1c, 512, 0, stream>>>(sim, mask, dg, dl);
  const int tot = Bc * N1Pc * N1Pc;
  gso_kernel<<<(tot + 255) / 256, 256, 0, stream>>>(sim, mask, dg, dl, gsoG,
                                                    gsoL);
  // GCN (global gso): two layers with residual
  gcn_layer_kernel<<<Bc * Lc, 256, 0, stream>>>(gsoG, all_nodes, gcn_w0,
                                                gcn_b0, bufT);
  gcn_layer_kernel<<<Bc * Lc, 256, 0, stream>>>(gsoG, bufT, gcn_w1, gcn_b1,
                                                bufGA);
  // GCN (local gso)
  gcn_layer_kernel<<<Bc * Lc, 256, 0, stream>>>(gsoL, all_nodes, gcn_w0,
                                                gcn_b0, bufT);
  gcn_layer_kernel<<<Bc * Lc, 256, 0, stream>>>(gsoL, bufT, gcn_w1, gcn_b1,
                                                bufLA);
  // gMLP over concat([ga, la]) -> gm (into bufT)
  const int Rb = (Bc * Lc * N1c) / 128;  // 2408
  gmlp_kernel<<<Rb, 256, 0, stream>>>(bufGA, bufLA, gmlp_w1, gmlp_b1, gmlp_w2,
                                      gmlp_b2, bufT);
  // node MLP on transposed gm -> output (B,L,64)
  nmlp_kernel<<<Bc * Lc, 256, 0, stream>>>(bufT, nmlp_w1, nmlp_b1, nmlp_w2,
                                           nmlp_b2, outp);
}